// NonLocalEncoder_17970143167165
// MI455X (gfx1250) — compile-verified
//
#include <hip/hip_runtime.h>
#include <math.h>

// ---------------------------------------------------------------------------
// NonLocalEncoder step for MI455X (gfx1250, wave32).
// All GEMMs run on V_WMMA_F32_16X16X4_F32 (full fp32 matrix pipe).
// Big GEMMs use 32x32 register blocking (2x2 WMMA sub-tiles per wave) to
// lift the wmma:vmem issue ratio (8 B/lane/wmma instead of 16).
// ---------------------------------------------------------------------------

typedef float v2f __attribute__((ext_vector_type(2)));
typedef float v8f __attribute__((ext_vector_type(8)));

constexpr int NM   = 768;   // edges == loops
constexpr int CDIM = 256;   // feature dim
constexpr int HID  = 64;    // head hidden dim
constexpr int IMGD = 1024;  // image dim
constexpr float EPS = 1e-4f;

#define WMMA_F32(a, b, c) \
  __builtin_amdgcn_wmma_f32_16x16x4_f32(false, (a), false, (b), (short)0, (c), false, false)

// ---------------------------------------------------------------------------
// Head: pred = sigmoid( relu(X @ W1 + b1) @ W2 + b2 )
// One block = 16 rows. 4 waves, each wave produces one 16x16 tile of the
// 16x64 hidden activation via fp32 WMMA (K loop = 256/4).
// ---------------------------------------------------------------------------
__global__ __launch_bounds__(128) void head_kernel(
    const float* __restrict__ X,    // [768,256]
    const float* __restrict__ W1,   // [256,64]
    const float* __restrict__ b1,   // [64]
    const float* __restrict__ W2,   // [64]
    const float* __restrict__ b2,   // [1]
    float* __restrict__ pred)       // [768]
{
  __shared__ float hid[16][HID + 4];
  const int wv   = threadIdx.x >> 5;
  const int lane = threadIdx.x & 31;
  const int half = lane >> 4;       // 0: lanes 0-15 (K=0,1), 1: lanes 16-31 (K=2,3)
  const int l16  = lane & 15;
  const int row0 = blockIdx.x * 16;
  const int col0 = wv * 16;

  v8f acc = {};
  for (int k = 0; k < CDIM; k += 4) {
    const int ka = k + (half ? 2 : 0);
    v2f a, b;
    a.x = X[(row0 + l16) * CDIM + ka];       // A: M=l16, K=ka   (VGPR0)
    a.y = X[(row0 + l16) * CDIM + ka + 1];   //    M=l16, K=ka+1 (VGPR1)
    b.x = W1[(ka) * HID + col0 + l16];       // B: K=ka,   N=l16
    b.y = W1[(ka + 1) * HID + col0 + l16];   //    K=ka+1, N=l16
    acc = WMMA_F32(a, b, acc);
  }
#pragma unroll
  for (int v = 0; v < 8; ++v) {
    const int r = v + half * 8;              // D layout: M = v (+8 for hi half)
    const int c = col0 + l16;
    hid[r][c] = fmaxf(acc[v] + b1[c], 0.0f);
  }
  __syncthreads();
  if (threadIdx.x < 16) {
    float s = b2[0];
#pragma unroll 8
    for (int c = 0; c < HID; ++c) s += hid[threadIdx.x][c] * W2[c];
    pred[row0 + threadIdx.x] = 1.0f / (1.0f + __expf(-s));
  }
}

// ---------------------------------------------------------------------------
// w[n,m] = mask[n,m] * |l0[n] - e0[m]|, plus row sums (w.sum over m).
// ---------------------------------------------------------------------------
__global__ __launch_bounds__(256) void w_kernel(
    const float* __restrict__ mask,   // [768,768]
    const float* __restrict__ e0,     // [768]
    const float* __restrict__ l0,     // [768]
    float* __restrict__ w,            // [768,768]
    float* __restrict__ rowsum)       // [768]
{
  __shared__ float red[256];
  const int n = blockIdx.x;
  const float ln = l0[n];
  float s = 0.0f;
  for (int m = threadIdx.x; m < NM; m += 256) {
    const float v = mask[n * NM + m] * fabsf(ln - e0[m]);
    w[n * NM + m] = v;
    s += v;
  }
  red[threadIdx.x] = s;
  __syncthreads();
  for (int off = 128; off > 0; off >>= 1) {
    if (threadIdx.x < off) red[threadIdx.x] += red[threadIdx.x + off];
    __syncthreads();
  }
  if (threadIdx.x == 0) rowsum[n] = red[0];
}

// Column sums of w (w.sum over n). One thread per column m; coalesced reads.
__global__ __launch_bounds__(256) void colsum_kernel(
    const float* __restrict__ w, float* __restrict__ colsum)
{
  const int m = blockIdx.x * 256 + threadIdx.x;
  float s = 0.0f;
  for (int n = 0; n < NM; ++n) s += w[n * NM + m];
  colsum[m] = s;
}

// ---------------------------------------------------------------------------
// Weighted aggregation GEMM, 32x32 register-blocked:
//   out[r,c] = (sum_k A(r,k) * X[k,c]) / max(div[r], EPS)
//   A(r,k) = transA ? Wm[k,r] : Wm[r,k]       (K = 768)
// One wave per 32x32 tile (2x2 WMMA sub-tiles); 24x8 = 192 tiles total.
// ---------------------------------------------------------------------------
__global__ __launch_bounds__(128) void wgemm_kernel(
    const float* __restrict__ Wm,    // [768,768]
    const float* __restrict__ X,     // [768,256]
    const float* __restrict__ div,   // [768]
    float* __restrict__ out,         // [768,256]
    int transA)
{
  const int wv   = threadIdx.x >> 5;
  const int lane = threadIdx.x & 31;
  const int half = lane >> 4;
  const int l16  = lane & 15;
  const int tile = blockIdx.x * 4 + wv;
  const int row0 = (tile >> 3) * 32;   // 24 row tiles of 32
  const int col0 = (tile & 7) * 32;    //  8 col tiles of 32

  v8f acc00 = {}, acc01 = {}, acc10 = {}, acc11 = {};
  for (int k = 0; k < NM; k += 4) {
    const int ka = k + (half ? 2 : 0);
    v2f a0, a1, b0, b1;
    if (transA) {
      a0.x = Wm[(ka) * NM + row0 + l16];
      a0.y = Wm[(ka + 1) * NM + row0 + l16];
      a1.x = Wm[(ka) * NM + row0 + 16 + l16];
      a1.y = Wm[(ka + 1) * NM + row0 + 16 + l16];
    } else {
      a0.x = Wm[(row0 + l16) * NM + ka];
      a0.y = Wm[(row0 + l16) * NM + ka + 1];
      a1.x = Wm[(row0 + 16 + l16) * NM + ka];
      a1.y = Wm[(row0 + 16 + l16) * NM + ka + 1];
    }
    b0.x = X[(ka) * CDIM + col0 + l16];
    b0.y = X[(ka + 1) * CDIM + col0 + l16];
    b1.x = X[(ka) * CDIM + col0 + 16 + l16];
    b1.y = X[(ka + 1) * CDIM + col0 + 16 + l16];
    acc00 = WMMA_F32(a0, b0, acc00);
    acc01 = WMMA_F32(a0, b1, acc01);
    acc10 = WMMA_F32(a1, b0, acc10);
    acc11 = WMMA_F32(a1, b1, acc11);
  }
#pragma unroll
  for (int v = 0; v < 8; ++v) {
    const int r0 = row0 + v + half * 8;
    const int r1 = r0 + 16;
    const float s0 = 1.0f / fmaxf(div[r0], EPS);
    const float s1 = 1.0f / fmaxf(div[r1], EPS);
    out[r0 * CDIM + col0 + l16]      = acc00[v] * s0;
    out[r0 * CDIM + col0 + 16 + l16] = acc01[v] * s0;
    out[r1 * CDIM + col0 + l16]      = acc10[v] * s1;
    out[r1 * CDIM + col0 + 16 + l16] = acc11[v] * s1;
  }
}

// ---------------------------------------------------------------------------
// Conflict max-pool: out[r,c] = max_j mask[r,j] * X[j,c]
// ---------------------------------------------------------------------------
__global__ __launch_bounds__(256) void conflict_max_kernel(
    const float* __restrict__ mask,  // [768,768]
    const float* __restrict__ X,     // [768,256]
    float* __restrict__ out)         // [768,256]
{
  const int r = blockIdx.x;
  const int c = threadIdx.x;
  float best = -INFINITY;
  for (int j = 0; j < NM; ++j)
    best = fmaxf(best, mask[r * NM + j] * X[j * CDIM + c]);
  out[r * CDIM + c] = best;
}

// ---------------------------------------------------------------------------
// Image contribution folded into the bias:
//   bias[c] = b3[c] + sum_k image_x[k] * W3[768 + k, c]
// Replaces 768x1024x256 GEMM work with one 1024x256 GEMV (-57% of FLOPs).
// ---------------------------------------------------------------------------
__global__ __launch_bounds__(256) void imgbias_kernel(
    const float* __restrict__ img,   // [1024]
    const float* __restrict__ W3,    // [1792,256]
    const float* __restrict__ b3,    // [256]
    float* __restrict__ bias)        // [256]
{
  const int c = threadIdx.x;
  float s = b3[c];
  for (int k = 0; k < IMGD; ++k) s += img[k] * W3[(3 * CDIM + k) * CDIM + c];
  bias[c] = s;
}

// ---------------------------------------------------------------------------
// Fused concat-GEMM, 32x32 register-blocked:
//   out = relu( [A0|A1|A2] @ W3[0:768,:] + bias )   (bias carries image part)
// ---------------------------------------------------------------------------
__global__ __launch_bounds__(128) void fused_gemm_kernel(
    const float* __restrict__ A0,    // [768,256] original features
    const float* __restrict__ A1,    // [768,256] weighted aggregation
    const float* __restrict__ A2,    // [768,256] conflict max-pool
    const float* __restrict__ W3,    // [1792,256], rows 0..767 used
    const float* __restrict__ bias,  // [256] (b3 + image contribution)
    float* __restrict__ out)         // [768,256]
{
  const int wv   = threadIdx.x >> 5;
  const int lane = threadIdx.x & 31;
  const int half = lane >> 4;
  const int l16  = lane & 15;
  const int tile = blockIdx.x * 4 + wv;
  const int row0 = (tile >> 3) * 32;
  const int col0 = (tile & 7) * 32;

  const float* As[3] = {A0, A1, A2};
  v8f acc00 = {}, acc01 = {}, acc10 = {}, acc11 = {};
  for (int k = 0; k < 3 * CDIM; k += 4) {
    const float* __restrict__ A = As[k >> 8];    // uniform chunk select
    const int kl = (k & (CDIM - 1)) + (half ? 2 : 0);
    const int kb = k + (half ? 2 : 0);
    v2f a0, a1, b0, b1;
    a0.x = A[(row0 + l16) * CDIM + kl];
    a0.y = A[(row0 + l16) * CDIM + kl + 1];
    a1.x = A[(row0 + 16 + l16) * CDIM + kl];
    a1.y = A[(row0 + 16 + l16) * CDIM + kl + 1];
    b0.x = W3[(kb) * CDIM + col0 + l16];
    b0.y = W3[(kb + 1) * CDIM + col0 + l16];
    b1.x = W3[(kb) * CDIM + col0 + 16 + l16];
    b1.y = W3[(kb + 1) * CDIM + col0 + 16 + l16];
    acc00 = WMMA_F32(a0, b0, acc00);
    acc01 = WMMA_F32(a0, b1, acc01);
    acc10 = WMMA_F32(a1, b0, acc10);
    acc11 = WMMA_F32(a1, b1, acc11);
  }
#pragma unroll
  for (int v = 0; v < 8; ++v) {
    const int r0 = row0 + v + half * 8;
    const int r1 = r0 + 16;
    const int c0 = col0 + l16;
    const int c1 = c0 + 16;
    out[r0 * CDIM + c0] = fmaxf(acc00[v] + bias[c0], 0.0f);
    out[r0 * CDIM + c1] = fmaxf(acc01[v] + bias[c1], 0.0f);
    out[r1 * CDIM + c0] = fmaxf(acc10[v] + bias[c0], 0.0f);
    out[r1 * CDIM + c1] = fmaxf(acc11[v] + bias[c1], 0.0f);
  }
}

// ---------------------------------------------------------------------------
// Launch
// ---------------------------------------------------------------------------
extern "C" void kernel_launch(void* const* d_in, const int* in_sizes, int n_in,
                              void* d_out, int out_size, void* d_ws, size_t ws_size,
                              hipStream_t stream) {
  const float* edge_x   = (const float*)d_in[0];
  const float* loop_x   = (const float*)d_in[1];
  const float* image_x  = (const float*)d_in[2];
  const float* le_mask  = (const float*)d_in[3];
  const float* ec_mask  = (const float*)d_in[4];
  const float* lc_mask  = (const float*)d_in[5];
  // d_in[6..8] unused by the reference computation
  const float* W_e3  = (const float*)d_in[9];
  const float* b_e3  = (const float*)d_in[10];
  const float* W_l3  = (const float*)d_in[11];
  const float* b_l3  = (const float*)d_in[12];
  const float* W_ep1 = (const float*)d_in[13];
  const float* b_ep1 = (const float*)d_in[14];
  const float* W_ep2 = (const float*)d_in[15];
  const float* b_ep2 = (const float*)d_in[16];
  const float* W_lp1 = (const float*)d_in[17];
  const float* b_lp1 = (const float*)d_in[18];
  const float* W_lp2 = (const float*)d_in[19];
  const float* b_lp2 = (const float*)d_in[20];

  // Output layout: [e0 | l0 | e1 | l1], 4 x 768 fp32
  float* e0 = (float*)d_out;
  float* l0 = e0 + NM;
  float* e1 = l0 + NM;
  float* l1 = e1 + NM;

  // Workspace layout (fp32), ~7.1 MB total
  float* ws = (float*)d_ws;
  size_t o = 0;
  float* w       = ws + o; o += (size_t)NM * NM;     // 589824
  float* rowsum  = ws + o; o += NM;
  float* colsum  = ws + o; o += NM;
  float* exl     = ws + o; o += (size_t)NM * CDIM;   // edge_x_from_loop
  float* lxe     = ws + o; o += (size_t)NM * CDIM;   // loop_x_from_edge
  float* exc     = ws + o; o += (size_t)NM * CDIM;   // edge_x_from_conflict
  float* lxc     = ws + o; o += (size_t)NM * CDIM;   // loop_x_from_conflict
  float* biasE   = ws + o; o += CDIM;                // b_e3 + image@W_e3
  float* biasL   = ws + o; o += CDIM;
  float* ex2     = ws + o; o += (size_t)NM * CDIM;   // edge_x2
  float* lx2     = ws + o; o += (size_t)NM * CDIM;   // loop_x2

  const int gemm_blocks = (NM / 32) * (CDIM / 32) / 4;   // 48 blocks x 4 waves

  // Stage 1: iteration-0 heads (WMMA fp32)
  head_kernel<<<NM / 16, 128, 0, stream>>>(edge_x, W_ep1, b_ep1, W_ep2, b_ep2, e0);
  head_kernel<<<NM / 16, 128, 0, stream>>>(loop_x, W_lp1, b_lp1, W_lp2, b_lp2, l0);

  // Stage 2: weights + row/col sums
  w_kernel<<<NM, 256, 0, stream>>>(le_mask, e0, l0, w, rowsum);
  colsum_kernel<<<NM / 256, 256, 0, stream>>>(w, colsum);

  // Stage 3: weighted aggregations (WMMA fp32, K=768, 32x32 blocked)
  wgemm_kernel<<<gemm_blocks, 128, 0, stream>>>(w, loop_x, colsum, exl, 1);
  wgemm_kernel<<<gemm_blocks, 128, 0, stream>>>(w, edge_x, rowsum, lxe, 0);

  // Stage 4: conflict max-pools (VALU)
  conflict_max_kernel<<<NM, 256, 0, stream>>>(ec_mask, edge_x, exc);
  conflict_max_kernel<<<NM, 256, 0, stream>>>(lc_mask, loop_x, lxc);

  // Stage 5: image GEMV folded into bias
  imgbias_kernel<<<1, 256, 0, stream>>>(image_x, W_e3, b_e3, biasE);
  imgbias_kernel<<<1, 256, 0, stream>>>(image_x, W_l3, b_l3, biasL);

  // Stage 6: fused concat-GEMM + ReLU (WMMA fp32, K=768, 32x32 blocked)
  fused_gemm_kernel<<<gemm_blocks, 128, 0, stream>>>(edge_x, exl, exc, W_e3, biasE, ex2);
  fused_gemm_kernel<<<gemm_blocks, 128, 0, stream>>>(loop_x, lxe, lxc, W_l3, biasL, lx2);

  // Stage 7: iteration-1 heads
  head_kernel<<<NM / 16, 128, 0, stream>>>(ex2, W_ep1, b_ep1, W_ep2, b_ep2, e1);
  head_kernel<<<NM / 16, 128, 0, stream>>>(lx2, W_lp1, b_lp1, W_lp2, b_lp2, l1);
}